// DeeperGCN_7756710936772
// MI455X (gfx1250) — compile-verified
//
#include <hip/hip_runtime.h>
#include <stdint.h>

#define NNODES 50000
#define NEDGES 800000
#define FIN    128
#define HDIM   64
#define H2     128
#define NLAYERS 4
#define LN_EPS  1e-5f
#define BN_EPS  1e-5f
#define GEN_EPS 1e-7f

typedef float v2f __attribute__((ext_vector_type(2)));
typedef float v8f __attribute__((ext_vector_type(8)));

// ---- order-preserving float <-> uint encoding for atomic max ----
__device__ __forceinline__ unsigned enc_ord(float f) {
    unsigned b = __float_as_uint(f);
    return (b & 0x80000000u) ? ~b : (b | 0x80000000u);
}
__device__ __forceinline__ float dec_ord(unsigned k) {
    unsigned b = (k & 0x80000000u) ? (k & 0x7fffffffu) : ~k;
    return __uint_as_float(b);
}
#define ENC_NEG_INF 0x007FFFFFu   // enc_ord(-inf)

// =====================================================================
// WMMA fp32 GEMM:  D[M,NCOL] = A[M,KDIM] @ W[KDIM,NCOL] + bias (+ D if RESID)
// One wave handles a 16-row tile and all NCOL columns (NCOL/16 accumulators).
// W is staged transposed into LDS so B fragments are contiguous float2 reads.
// =====================================================================
template<int KDIM, int NCOL, bool RESID>
__global__ __launch_bounds__(256) void gemm_wmma_kernel(
    const float* __restrict__ A, const float* __restrict__ W,
    const float* __restrict__ bias, float* __restrict__ D, int M)
{
    __shared__ float lds_wt[NCOL * KDIM];   // transposed: [col][k]
    __shared__ float lds_b[NCOL];

    const int tid = threadIdx.x;
    for (int i = tid; i < KDIM * NCOL; i += 256) {
        int k = i / NCOL, c = i % NCOL;
        lds_wt[c * KDIM + k] = W[i];
    }
    for (int i = tid; i < NCOL; i += 256) lds_b[i] = bias[i];
    __syncthreads();

    const int wave = tid >> 5;
    const int lane = tid & 31;
    const int half = lane >> 4;     // 0: lanes 0-15 (K pair k,k+1), 1: lanes 16-31 (k+2,k+3)
    const int l16  = lane & 15;
    const int rowBase = (blockIdx.x * 8 + wave) * 16;

    int row  = rowBase + l16;
    int rowc = (row < M) ? row : (M - 1);
    const float* arow = A + (size_t)rowc * KDIM;

    constexpr int NT = NCOL / 16;
    v8f acc[NT] = {};

    #pragma unroll
    for (int k = 0; k < KDIM; k += 4) {
        // A fragment: row = rowBase + l16, K = k + 2*half + {0,1}
        v2f a = *(const v2f*)(arow + k + 2 * half);
        #pragma unroll
        for (int j = 0; j < NT; ++j) {
            int col = j * 16 + l16;
            // B fragment: N = col, K = k + 2*half + {0,1}
            v2f b = *(const v2f*)(&lds_wt[col * KDIM + k + 2 * half]);
            acc[j] = __builtin_amdgcn_wmma_f32_16x16x4_f32(
                false, a, false, b, (short)0, acc[j], false, false);
        }
    }

    // D layout: acc[j] element i -> row rowBase + i + 8*half, col j*16 + l16
    #pragma unroll
    for (int j = 0; j < NT; ++j) {
        int col = j * 16 + l16;
        float bv = lds_b[col];
        #pragma unroll
        for (int i = 0; i < 8; ++i) {
            int r = rowBase + 8 * half + i;
            if (r < M) {
                size_t o = (size_t)r * NCOL + col;
                float v = acc[j][i] + bv;
                if (RESID) v += D[o];
                D[o] = v;
            }
        }
    }
}

// =====================================================================
// LayerNorm (+ReLU) over C channels, one wave per node, wave32 shuffles.
// =====================================================================
template<int C>
__global__ __launch_bounds__(256) void ln_relu_kernel(
    const float* __restrict__ X, const float* __restrict__ g,
    const float* __restrict__ b, float* __restrict__ Y, int n)
{
    constexpr int V = C / 32;
    const int wave = threadIdx.x >> 5;
    const int lane = threadIdx.x & 31;
    const int node = blockIdx.x * 8 + wave;
    if (node >= n) return;

    const float* x = X + (size_t)node * C + lane * V;
    float v[V];
    float s = 0.f;
    #pragma unroll
    for (int i = 0; i < V; ++i) { v[i] = x[i]; s += v[i]; }
    #pragma unroll
    for (int m = 16; m >= 1; m >>= 1) s += __shfl_xor(s, m, 32);
    float mu = s * (1.0f / C);

    float q = 0.f;
    #pragma unroll
    for (int i = 0; i < V; ++i) { float d = v[i] - mu; q += d * d; }
    #pragma unroll
    for (int m = 16; m >= 1; m >>= 1) q += __shfl_xor(q, m, 32);
    float inv = rsqrtf(q * (1.0f / C) + LN_EPS);

    float* y = Y + (size_t)node * C + lane * V;
    #pragma unroll
    for (int i = 0; i < V; ++i) {
        int c = lane * V + i;
        float o = (v[i] - mu) * inv * g[c] + b[c];
        y[i] = fmaxf(o, 0.f);
    }
}

// =====================================================================
// Edge phase
// =====================================================================
__global__ __launch_bounds__(256) void init_agg_kernel(
    unsigned* __restrict__ MX, float* __restrict__ S, float* __restrict__ Wm)
{
    int i = blockIdx.x * 256 + threadIdx.x;
    if (i < NNODES * HDIM) { MX[i] = ENC_NEG_INF; S[i] = 0.f; Wm[i] = 0.f; }
}

__global__ __launch_bounds__(256) void edge_max_kernel(
    const float* __restrict__ R, const int* __restrict__ src,
    const int* __restrict__ dst, const float* __restrict__ tptr, int tidx,
    unsigned* __restrict__ MX)
{
    int idx = blockIdx.x * 256 + threadIdx.x;        // E*16 threads, 4 channels each
    if (idx >= NEDGES * 16) return;
    int e = idx >> 4;
    int c = (idx & 15) * 4;
    float t = tptr[tidx];
    int sn = src[e], dn = dst[e];
    float4 xv = *(const float4*)(R + (size_t)sn * HDIM + c);
    unsigned* base = MX + (size_t)dn * HDIM + c;
    atomicMax(base + 0, enc_ord((fmaxf(xv.x, 0.f) + GEN_EPS) * t));
    atomicMax(base + 1, enc_ord((fmaxf(xv.y, 0.f) + GEN_EPS) * t));
    atomicMax(base + 2, enc_ord((fmaxf(xv.z, 0.f) + GEN_EPS) * t));
    atomicMax(base + 3, enc_ord((fmaxf(xv.w, 0.f) + GEN_EPS) * t));
}

__global__ __launch_bounds__(256) void edge_sum_kernel(
    const float* __restrict__ R, const int* __restrict__ src,
    const int* __restrict__ dst, const float* __restrict__ tptr, int tidx,
    const unsigned* __restrict__ MX, float* __restrict__ S, float* __restrict__ Wm)
{
    int idx = blockIdx.x * 256 + threadIdx.x;
    if (idx >= NEDGES * 16) return;
    int e = idx >> 4;
    int c = (idx & 15) * 4;
    float t = tptr[tidx];
    int sn = src[e], dn = dst[e];
    float4 xv = *(const float4*)(R + (size_t)sn * HDIM + c);
    size_t off = (size_t)dn * HDIM + c;
    float msg[4] = { fmaxf(xv.x,0.f)+GEN_EPS, fmaxf(xv.y,0.f)+GEN_EPS,
                     fmaxf(xv.z,0.f)+GEN_EPS, fmaxf(xv.w,0.f)+GEN_EPS };
    #pragma unroll
    for (int i = 0; i < 4; ++i) {
        float mx = dec_ord(MX[off + i]);
        float ee = __expf(msg[i] * t - mx);
        atomicAdd(S  + off + i, ee);
        atomicAdd(Wm + off + i, ee * msg[i]);
    }
}

// y = wsum / max(s,1e-16) + x   (in place on R: R becomes the MLP input)
__global__ __launch_bounds__(256) void combine_kernel(
    float* __restrict__ R, const float* __restrict__ S, const float* __restrict__ Wm)
{
    int i = blockIdx.x * 256 + threadIdx.x;
    if (i < NNODES * HDIM) R[i] = Wm[i] / fmaxf(S[i], 1e-16f) + R[i];
}

// eval BatchNorm (running stats mean=0,var=1) + ReLU, in place on U [N,128]
__global__ __launch_bounds__(256) void bn_relu_kernel(
    float* __restrict__ U, const float* __restrict__ g, const float* __restrict__ b)
{
    int i = blockIdx.x * 256 + threadIdx.x;
    if (i < NNODES * H2) {
        int c = i & (H2 - 1);
        float sc = g[c] * rsqrtf(1.0f + BN_EPS);
        U[i] = fmaxf(U[i] * sc + b[c], 0.f);
    }
}

// =====================================================================
extern "C" void kernel_launch(void* const* d_in, const int* in_sizes, int n_in,
                              void* d_out, int out_size, void* d_ws, size_t ws_size,
                              hipStream_t stream)
{
    (void)in_sizes; (void)n_in; (void)out_size; (void)ws_size;

    const float* x        = (const float*)d_in[0];
    const int*   ei       = (const int*)  d_in[1];
    const float* enc_W    = (const float*)d_in[2];
    const float* enc_b    = (const float*)d_in[3];
    const float* conv_t   = (const float*)d_in[4];
    const float* conv_W1  = (const float*)d_in[5];
    const float* conv_b1  = (const float*)d_in[6];
    const float* conv_lng = (const float*)d_in[7];
    const float* conv_lnb = (const float*)d_in[8];
    const float* conv_W2  = (const float*)d_in[9];
    const float* conv_b2  = (const float*)d_in[10];
    const float* block_lng= (const float*)d_in[11];
    const float* block_lnb= (const float*)d_in[12];
    const float* fin_t    = (const float*)d_in[13];
    const float* fin_W1   = (const float*)d_in[14];
    const float* fin_b1   = (const float*)d_in[15];
    const float* fin_bng  = (const float*)d_in[16];
    const float* fin_bnb  = (const float*)d_in[17];
    const float* fin_W2   = (const float*)d_in[18];
    const float* fin_b2   = (const float*)d_in[19];

    const int* src = ei;
    const int* dst = ei + NEDGES;

    // workspace layout (fp32): h[N*64] | r[N*64] | u[N*128] | mx[N*64]
    float*    h  = (float*)d_ws;
    float*    r  = h + (size_t)NNODES * HDIM;
    float*    u  = r + (size_t)NNODES * HDIM;
    unsigned* mx = (unsigned*)(u + (size_t)NNODES * H2);
    float*    S  = u;                              // alias: u dead during edge phase
    float*    Wm = u + (size_t)NNODES * HDIM;

    const int GEMM_GRID = (NNODES + 127) / 128;    // 8 waves/block * 16 rows/wave
    const int EDGE_GRID = (NEDGES * 16 + 255) / 256;
    const int NODE_GRID = (NNODES * HDIM + 255) / 256;
    const int LN_GRID   = (NNODES + 7) / 8;

    auto genconv = [&](float* xin, float* dest, bool resid, bool use_bn,
                       const float* tp, int ti,
                       const float* W1, const float* b1,
                       const float* ng, const float* nb,
                       const float* W2, const float* b2)
    {
        init_agg_kernel<<<NODE_GRID, 256, 0, stream>>>(mx, S, Wm);
        edge_max_kernel<<<EDGE_GRID, 256, 0, stream>>>(xin, src, dst, tp, ti, mx);
        edge_sum_kernel<<<EDGE_GRID, 256, 0, stream>>>(xin, src, dst, tp, ti, mx, S, Wm);
        combine_kernel <<<NODE_GRID, 256, 0, stream>>>(xin, S, Wm);
        gemm_wmma_kernel<HDIM, H2, false><<<GEMM_GRID, 256, 0, stream>>>(xin, W1, b1, u, NNODES);
        if (use_bn)
            bn_relu_kernel<<<(NNODES * H2 + 255) / 256, 256, 0, stream>>>(u, ng, nb);
        else
            ln_relu_kernel<H2><<<LN_GRID, 256, 0, stream>>>(u, ng, nb, u, NNODES);
        if (resid)
            gemm_wmma_kernel<H2, HDIM, true ><<<GEMM_GRID, 256, 0, stream>>>(u, W2, b2, dest, NNODES);
        else
            gemm_wmma_kernel<H2, HDIM, false><<<GEMM_GRID, 256, 0, stream>>>(u, W2, b2, dest, NNODES);
    };

    // encoder: h = x @ enc_W + enc_b
    gemm_wmma_kernel<FIN, HDIM, false><<<GEMM_GRID, 256, 0, stream>>>(x, enc_W, enc_b, h, NNODES);

    // layer 0: h = genconv(h)
    genconv(h, h, /*resid=*/false, /*bn=*/false, conv_t, 0,
            conv_W1, conv_b1, conv_lng, conv_lnb, conv_W2, conv_b2);

    // res+ blocks 1..3: h = h + genconv(relu(LN(h)))
    for (int i = 1; i < NLAYERS; ++i) {
        ln_relu_kernel<HDIM><<<LN_GRID, 256, 0, stream>>>(
            h, block_lng + i * HDIM, block_lnb + i * HDIM, r, NNODES);
        genconv(r, h, /*resid=*/true, /*bn=*/false, conv_t, i,
                conv_W1 + (size_t)i * HDIM * H2, conv_b1 + (size_t)i * H2,
                conv_lng + (size_t)i * H2,       conv_lnb + (size_t)i * H2,
                conv_W2 + (size_t)i * H2 * HDIM, conv_b2 + (size_t)i * HDIM);
    }

    // final: out = genconv_bn(relu(LN(h, block[0])))
    ln_relu_kernel<HDIM><<<LN_GRID, 256, 0, stream>>>(
        h, block_lng, block_lnb, r, NNODES);
    genconv(r, (float*)d_out, /*resid=*/false, /*bn=*/true, fin_t, 0,
            fin_W1, fin_b1, fin_bng, fin_bnb, fin_W2, fin_b2);
}